// GNNModel_72327249265173
// MI455X (gfx1250) — compile-verified
//
#include <hip/hip_runtime.h>
#include <hip/hip_bf16.h>

typedef __attribute__((ext_vector_type(2))) float v2f;
typedef __attribute__((ext_vector_type(8))) float v8f;

#define NNODES 50000
#define NEDGES 800000
#define INDIM  128
#define HID    64
#define NGRAPH 256

// ---------------- degree / dinv ----------------
__global__ void deg_kernel(const int* __restrict__ dst, float* __restrict__ deg, int E) {
    int tid = blockIdx.x * blockDim.x + threadIdx.x;
    if (tid < E) atomicAdd(deg + dst[tid], 1.0f);
}

__global__ void dinv_kernel(float* __restrict__ deg, int n) {
    int tid = blockIdx.x * blockDim.x + threadIdx.x;
    if (tid < n) deg[tid] = rsqrtf(deg[tid] + 1.0f);  // +1 = self-loop, always > 0
}

// ---------------- WMMA GEMM: P[M x 64] = (A[M x K] @ W[K x 64]) * dinv[row] ----------------
// One wave computes a 16x64 tile using V_WMMA_F32_16X16X4_F32 (exact fp32 math).
// A frag (16x4):  lane = M%16 + 16*(K>=2); vgpr0/1 = consecutive K.
// B frag (4x16):  vgpr v, lane-half h hold row K = v + 2h, col = lane%16.
// C/D (16x16):    vgpr v, lane l: row = v + 8*(l>=16), col = l%16.
template <int K>
__global__ void gemm_wmma_scaled(const float* __restrict__ A, const float* __restrict__ W,
                                 const float* __restrict__ dinv, float* __restrict__ P, int M) {
    const int wave = (int)((blockIdx.x * (unsigned)blockDim.x + threadIdx.x) >> 5);
    const int lane = threadIdx.x & 31;
    const int row0 = wave << 4;
    if (row0 >= M) return;  // wave-uniform: EXEC stays all-ones for active waves

    const int mrow = row0 + (lane & 15);
    const int kh   = (lane >> 4) << 1;   // 0 or 2 (K half selected by upper lane group)
    const int ncol = lane & 15;

    v8f acc0 = {}, acc1 = {}, acc2 = {}, acc3 = {};
    for (int k = 0; k < K; k += 4) {
        v2f a = *(const v2f*)(A + (size_t)mrow * K + (k + kh));
        const float* wp = W + (size_t)(k + kh) * HID + ncol;
        v2f b0 = { wp[0],  wp[HID] };
        v2f b1 = { wp[16], wp[16 + HID] };
        v2f b2 = { wp[32], wp[32 + HID] };
        v2f b3 = { wp[48], wp[48 + HID] };
        acc0 = __builtin_amdgcn_wmma_f32_16x16x4_f32(false, a, false, b0, (short)0, acc0, false, false);
        acc1 = __builtin_amdgcn_wmma_f32_16x16x4_f32(false, a, false, b1, (short)0, acc1, false, false);
        acc2 = __builtin_amdgcn_wmma_f32_16x16x4_f32(false, a, false, b2, (short)0, acc2, false, false);
        acc3 = __builtin_amdgcn_wmma_f32_16x16x4_f32(false, a, false, b3, (short)0, acc3, false, false);
    }

    const int rbase = row0 + ((lane >> 4) << 3);
#pragma unroll
    for (int v = 0; v < 8; ++v) {
        float d = dinv[rbase + v];
        float* pp = P + (size_t)(rbase + v) * HID + ncol;
        pp[0]  = acc0[v] * d;
        pp[16] = acc1[v] * d;
        pp[32] = acc2[v] * d;
        pp[48] = acc3[v] * d;
    }
}

// ---------------- edge scatter-add: acc[dst] += p[src] (64 floats/edge) ----------------
__global__ void scatter_add(const float* __restrict__ p, const int* __restrict__ src,
                            const int* __restrict__ dst, float* __restrict__ acc) {
    long long tid = (long long)blockIdx.x * blockDim.x + threadIdx.x;
    if (tid >= (long long)NEDGES * HID) return;
    int e = (int)(tid >> 6);
    int c = (int)(tid & 63);
    atomicAdd(acc + (size_t)dst[e] * HID + c, p[(size_t)src[e] * HID + c]);
}

// ---------------- h = relu(dinv*(acc + p_self) + bias), in place into acc ----------------
__global__ void finalize_kernel(float* __restrict__ acc, const float* __restrict__ p,
                                const float* __restrict__ dinv, const float* __restrict__ bias) {
    long long tid = (long long)blockIdx.x * blockDim.x + threadIdx.x;
    if (tid >= (long long)NNODES * HID) return;
    int i = (int)(tid >> 6);
    int c = (int)(tid & 63);
    float v = dinv[i] * (acc[tid] + p[tid]) + bias[c];
    acc[tid] = fmaxf(v, 0.0f);
}

// ---------------- mean-pool accumulation ----------------
__global__ void pool_kernel(const float* __restrict__ h, const int* __restrict__ batch,
                            float* __restrict__ pooled, float* __restrict__ counts) {
    long long tid = (long long)blockIdx.x * blockDim.x + threadIdx.x;
    if (tid >= (long long)NNODES * HID) return;
    int i = (int)(tid >> 6);
    int c = (int)(tid & 63);
    int g = batch[i];
    atomicAdd(pooled + (size_t)g * HID + c, h[tid]);
    if (c == 0) atomicAdd(counts + g, 1.0f);
}

// ---------------- head: out[g] = (pooled[g]/max(cnt,1)) . Wl + bl ----------------
__global__ void head_kernel(const float* __restrict__ pooled, const float* __restrict__ counts,
                            const float* __restrict__ Wl, const float* __restrict__ bl,
                            float* __restrict__ out) {
    int g = threadIdx.x;  // one block of 256
    float inv = 1.0f / fmaxf(counts[g], 1.0f);
    float s = 0.0f;
#pragma unroll
    for (int c = 0; c < HID; ++c) s += pooled[(size_t)g * HID + c] * Wl[c];
    out[g] = s * inv + bl[0];
}

extern "C" void kernel_launch(void* const* d_in, const int* in_sizes, int n_in,
                              void* d_out, int out_size, void* d_ws, size_t ws_size,
                              hipStream_t stream) {
    const float* x   = (const float*)d_in[0];   // [50000,128]
    const float* W1  = (const float*)d_in[1];   // [128,64]
    const float* b1  = (const float*)d_in[2];   // [64]
    const float* W2  = (const float*)d_in[3];   // [64,64]
    const float* b2  = (const float*)d_in[4];   // [64]
    const float* Wl  = (const float*)d_in[5];   // [64]
    const float* bl  = (const float*)d_in[6];   // [1]
    const int*   ei  = (const int*)d_in[7];     // [2,800000] flat
    const int*   bat = (const int*)d_in[8];     // [50000]
    const int* src = ei;
    const int* dst = ei + NEDGES;
    float* out = (float*)d_out;

    // workspace carve-up (~26 MB of floats)
    float* ws     = (float*)d_ws;
    float* dinv   = ws;                                  // 50000 (deg -> dinv in place)
    float* bufA   = ws + 50048;                          // 50000*64  (p = (h*dinv))
    float* bufB   = bufA + (size_t)NNODES * HID;         // 50000*64  (acc -> h in place)
    float* pooled = bufB + (size_t)NNODES * HID;         // 256*64
    float* counts = pooled + (size_t)NGRAPH * HID;       // 256

    const int mtiles = NNODES / 16;                      // 3125 exactly
    const int gemmBlocks = (mtiles + 3) / 4;             // 4 waves (128 thr) per block
    const long long eWork = (long long)NEDGES * HID;     // 51.2M
    const int eBlocks = (int)((eWork + 255) / 256);
    const long long nWork = (long long)NNODES * HID;     // 3.2M
    const int nBlocks = (int)((nWork + 255) / 256);

    // degrees -> dinv
    hipMemsetAsync(dinv, 0, NNODES * sizeof(float), stream);
    deg_kernel<<<(NEDGES + 255) / 256, 256, 0, stream>>>(dst, dinv, NEDGES);
    dinv_kernel<<<(NNODES + 255) / 256, 256, 0, stream>>>(dinv, NNODES);

    // ---- layer 1 ----
    gemm_wmma_scaled<INDIM><<<gemmBlocks, 128, 0, stream>>>(x, W1, dinv, bufA, NNODES);
    hipMemsetAsync(bufB, 0, (size_t)NNODES * HID * sizeof(float), stream);
    scatter_add<<<eBlocks, 256, 0, stream>>>(bufA, src, dst, bufB);
    finalize_kernel<<<nBlocks, 256, 0, stream>>>(bufB, bufA, dinv, b1);

    // ---- layer 2 ----
    gemm_wmma_scaled<HID><<<gemmBlocks, 128, 0, stream>>>(bufB, W2, dinv, bufA, NNODES);
    hipMemsetAsync(bufB, 0, (size_t)NNODES * HID * sizeof(float), stream);
    scatter_add<<<eBlocks, 256, 0, stream>>>(bufA, src, dst, bufB);
    finalize_kernel<<<nBlocks, 256, 0, stream>>>(bufB, bufA, dinv, b2);

    // ---- mean pool + linear head ----
    hipMemsetAsync(pooled, 0, ((size_t)NGRAPH * HID + NGRAPH) * sizeof(float), stream);
    pool_kernel<<<nBlocks, 256, 0, stream>>>(bufB, bat, pooled, counts);
    head_kernel<<<1, NGRAPH, 0, stream>>>(pooled, counts, Wl, bl, out);
}